// SE_9096740733112
// MI455X (gfx1250) — compile-verified
//
#include <hip/hip_runtime.h>
#include <math.h>

// ---------------------------------------------------------------------------
// SE-gate over graph batches (segment mean pool -> 2-layer MLP gate -> scale)
// MI455X / gfx1250, wave32.  Bandwidth-bound: ~768MB traffic => ~33us floor.
// MLP done with V_WMMA_F32_16X16X4_F32 on a single wave (fp32 fidelity).
// ---------------------------------------------------------------------------

typedef __attribute__((ext_vector_type(2))) float v2f;
typedef __attribute__((ext_vector_type(4))) float v4f;
typedef __attribute__((ext_vector_type(8))) float v8f;

#define SE_C 128          // channels
#define SE_B 32           // graphs (batch_num)
#define ROWS_PER_BLOCK 512

// ---------------------------------------------------------------------------
// Kernel 0: zero the accumulator region of the workspace (sums + counts).
// ---------------------------------------------------------------------------
__global__ __launch_bounds__(256) void se_zero_kernel(float* __restrict__ p, int n) {
    int i = blockIdx.x * 256 + threadIdx.x;
    if (i < n) p[i] = 0.0f;
}

// ---------------------------------------------------------------------------
// Kernel 1: segment sums + counts.  batch is sorted, so each block's 512-row
// chunk spans at most a couple of segments.  Thread t owns column t =>
// LDS accumulation needs no atomics; only the touched segment range
// [batch[r0], batch[r1-1]] is flushed to global with atomics.
// ---------------------------------------------------------------------------
__global__ __launch_bounds__(128) void se_pool_kernel(
    const float* __restrict__ x, const int* __restrict__ batch, int n,
    float* __restrict__ gsums, float* __restrict__ gcounts) {

    __shared__ float lsum[SE_B][SE_C];
    __shared__ float lcnt[SE_B];
    __shared__ int   lbatch[ROWS_PER_BLOCK];

    const int c  = threadIdx.x;                      // column 0..127
    const int r0 = blockIdx.x * ROWS_PER_BLOCK;
    if (r0 >= n) return;
    const int r1 = min(r0 + ROWS_PER_BLOCK, n);
    const int nrows = r1 - r0;

    // stage batch ids for this chunk (2KB) + zero the LDS accumulators
    for (int i = c; i < ROWS_PER_BLOCK; i += 128)
        lbatch[i] = (r0 + i < n) ? batch[r0 + i] : -1;
    #pragma unroll
    for (int b = 0; b < SE_B; ++b) lsum[b][c] = 0.0f;
    if (c < SE_B) lcnt[c] = 0.0f;
    __syncthreads();

    int   bcur = lbatch[0];
    float acc  = 0.0f;
    float cnt  = 0.0f;
    for (int r = 0; r < nrows; ++r) {
        int b = lbatch[r];
        if (b != bcur) {                              // rare: segment boundary
            lsum[bcur][c] += acc;
            if (c == 0) lcnt[bcur] += cnt;
            acc = 0.0f; cnt = 0.0f; bcur = b;
        }
        // stream x: 4 waves x 128B contiguous per row, non-temporal (256MB stream)
        acc += __builtin_nontemporal_load(&x[(long long)(r0 + r) * SE_C + c]);
        cnt += 1.0f;
    }
    lsum[bcur][c] += acc;
    if (c == 0) lcnt[bcur] += cnt;
    __syncthreads();

    const int bmin = lbatch[0];
    const int bmax = lbatch[nrows - 1];
    for (int b = bmin; b <= bmax; ++b) {              // typically 1-2 segments
        atomicAdd(&gsums[b * SE_C + c], lsum[b][c]);
        if (c == 0) atomicAdd(&gcounts[b], lcnt[b]);
    }
}

// ---------------------------------------------------------------------------
// Kernel 2: single wave32 computes  attn = sigmoid(relu(avg@W1+b1)@W2+b2)
// via V_WMMA_F32_16X16X4_F32.
//   A 16x4 f32 frag (2 VGPR): lanes 0-15 K=0/1, lanes 16-31 K=2/3 (ISA 7.12.2)
//   B 4x16 f32 frag (2 VGPR): VGPR0 K=0|K=2 across lane halves, VGPR1 K=1|K=3
//   C/D 16x16 f32 (8 VGPR):   VGPR r -> M=r (lanes 0-15), M=r+8 (lanes 16-31)
// EXEC must be all ones: blockDim = 32, no divergence around WMMA.
// ---------------------------------------------------------------------------
__global__ __launch_bounds__(32) void se_mlp_kernel(
    const float* __restrict__ sums, const float* __restrict__ counts,
    const float* __restrict__ W1, const float* __restrict__ b1,
    const float* __restrict__ W2, const float* __restrict__ b2,
    float* __restrict__ attn) {

    __shared__ float avgL[SE_B * SE_C];   // 32x128
    __shared__ float hL[SE_B * SE_B];     // 32x32

    const int lane = threadIdx.x;
    const int m    = lane & 15;           // row / col within tile
    const int half = lane >> 4;           // 0 | 1
    const int koff = half * 2;            // K sub-offset for A/B frags

    // avg = sums / max(counts, 1)
    for (int i = lane; i < SE_B * SE_C; i += 32) {
        float cnt = counts[i >> 7];
        avgL[i] = sums[i] / fmaxf(cnt, 1.0f);
    }
    __syncthreads();

    // GEMM1: H = relu(avg[32,128] @ W1[128,32] + b1)   (2x2 tiles, 32 k-steps)
    #pragma unroll
    for (int Mt = 0; Mt < 2; ++Mt) {
        #pragma unroll
        for (int Nt = 0; Nt < 2; ++Nt) {
            v8f acc = {};
            for (int k4 = 0; k4 < SE_C; k4 += 4) {
                v2f a, b;
                a.x = avgL[(Mt * 16 + m) * SE_C + k4 + koff];
                a.y = avgL[(Mt * 16 + m) * SE_C + k4 + koff + 1];
                b.x = W1[(k4 + koff) * SE_B + Nt * 16 + m];
                b.y = W1[(k4 + koff + 1) * SE_B + Nt * 16 + m];
                acc = __builtin_amdgcn_wmma_f32_16x16x4_f32(
                    false, a, false, b, (short)0, acc, false, false);
            }
            const int col  = Nt * 16 + m;
            const float bb = b1[col];
            #pragma unroll
            for (int r = 0; r < 8; ++r) {
                int row = Mt * 16 + r + 8 * half;
                float h = acc[r] + bb;
                hL[row * SE_B + col] = h > 0.0f ? h : 0.0f;
            }
        }
    }
    __syncthreads();

    // GEMM2: attn = sigmoid(H[32,32] @ W2[32,128] + b2) (2x8 tiles, 8 k-steps)
    #pragma unroll
    for (int Mt = 0; Mt < 2; ++Mt) {
        for (int Nt = 0; Nt < 8; ++Nt) {
            v8f acc = {};
            #pragma unroll
            for (int k4 = 0; k4 < SE_B; k4 += 4) {
                v2f a, b;
                a.x = hL[(Mt * 16 + m) * SE_B + k4 + koff];
                a.y = hL[(Mt * 16 + m) * SE_B + k4 + koff + 1];
                b.x = W2[(k4 + koff) * SE_C + Nt * 16 + m];
                b.y = W2[(k4 + koff + 1) * SE_C + Nt * 16 + m];
                acc = __builtin_amdgcn_wmma_f32_16x16x4_f32(
                    false, a, false, b, (short)0, acc, false, false);
            }
            const int col  = Nt * 16 + m;
            const float bb = b2[col];
            #pragma unroll
            for (int r = 0; r < 8; ++r) {
                int row = Mt * 16 + r + 8 * half;
                float v = acc[r] + bb;
                attn[row * SE_C + col] = 1.0f / (1.0f + __expf(-v));
            }
        }
    }
}

// ---------------------------------------------------------------------------
// Kernel 3: out = x * attn[batch]  (float4-vectorized stream; attn in LDS;
// non-temporal for the 512MB x/out stream so it doesn't churn L2).
// ---------------------------------------------------------------------------
__global__ __launch_bounds__(256) void se_scale_kernel(
    const float* __restrict__ x, const int* __restrict__ batch,
    const float* __restrict__ attn, float* __restrict__ out, int n) {

    __shared__ v4f lattn[SE_B * SE_C / 4];            // 16KB
    const int t = threadIdx.x;
    const v4f* attn4 = (const v4f*)attn;
    for (int i = t; i < SE_B * SE_C / 4; i += 256) lattn[i] = attn4[i];
    __syncthreads();

    const v4f* x4   = (const v4f*)x;
    v4f*       out4 = (v4f*)out;
    const long long total  = (long long)n * (SE_C / 4);
    const long long stride = (long long)gridDim.x * 256;
    for (long long i = (long long)blockIdx.x * 256 + t; i < total; i += stride) {
        int row = (int)(i >> 5);                      // C/4 == 32
        int cg  = (int)(i & 31);
        int b   = batch[row];                         // broadcast within 32 lanes
        v4f g   = lattn[b * 32 + cg];
        v4f v   = __builtin_nontemporal_load(&x4[i]);
        v.x *= g.x; v.y *= g.y; v.z *= g.z; v.w *= g.w;
        __builtin_nontemporal_store(v, &out4[i]);
    }
}

// ---------------------------------------------------------------------------
// Launch: zero -> pool -> MLP (WMMA) -> scale, all on `stream`.
// Workspace layout (floats): [0, 4096) sums, [4096, 4128) counts,
//                            [4128, 8224) attn.
// ---------------------------------------------------------------------------
extern "C" void kernel_launch(void* const* d_in, const int* in_sizes, int n_in,
                              void* d_out, int out_size, void* d_ws, size_t ws_size,
                              hipStream_t stream) {
    const float* x     = (const float*)d_in[0];
    const int*   batch = (const int*)d_in[1];   // harness ABI: integer -> int32
    // d_in[2] = batch_num scalar (compile-time SE_B=32 per reference)
    const float* W1 = (const float*)d_in[3];
    const float* b1 = (const float*)d_in[4];
    const float* W2 = (const float*)d_in[5];
    const float* b2 = (const float*)d_in[6];

    const int n = in_sizes[0] / SE_C;           // 500000 nodes

    float* sums   = (float*)d_ws;               // [32,128]
    float* counts = sums + SE_B * SE_C;         // [32]
    float* attn   = counts + SE_B;              // [32,128] (offset 16512B, 16B-aligned)

    const int zn = SE_B * SE_C + SE_B;
    se_zero_kernel<<<(zn + 255) / 256, 256, 0, stream>>>(sums, zn);

    const int nblk = (n + ROWS_PER_BLOCK - 1) / ROWS_PER_BLOCK;
    se_pool_kernel<<<nblk, 128, 0, stream>>>(x, batch, n, sums, counts);

    se_mlp_kernel<<<1, 32, 0, stream>>>(sums, counts, W1, b1, W2, b2, attn);

    se_scale_kernel<<<4096, 256, 0, stream>>>(x, batch, attn, (float*)d_out, n);
}